// HyperbolicTripletLoss_16887811408098
// MI455X (gfx1250) — compile-verified
//
#include <hip/hip_runtime.h>
#include <math.h>

typedef float v2f __attribute__((ext_vector_type(2)));
typedef float v8f __attribute__((ext_vector_type(8)));

#define NB 4096
#define ND 128
#define EPSF 1e-7f
#define LDS_STRIDE 132   // 128 + 4 pad: bank = (4*row + col) & 63 -> conflict-free

__device__ __forceinline__ float wave_red_addf(float v) {
  #pragma unroll
  for (int m = 1; m < 32; m <<= 1) v += __shfl_xor(v, m, 32);
  return v;
}
__device__ __forceinline__ unsigned wave_red_addu(unsigned v) {
  #pragma unroll
  for (int m = 1; m < 32; m <<= 1) v += __shfl_xor(v, m, 32);
  return v;
}

// Poincare distance from dot product and squared norms (c = 1)
__device__ __forceinline__ float pdist(float dot, float si, float sj) {
  float diff = fmaxf(si + sj - 2.0f * dot, 0.0f);
  float den  = fmaxf((1.0f - si) * (1.0f - sj), EPSF);
  float arg  = 1.0f + 2.0f * diff / den;
  return acoshf(fmaxf(arg, 1.0f));
}

__device__ __forceinline__ void load_atile(v2f* areg, const float* __restrict__ X,
                                           int i0, int lane) {
  const int row  = i0 + (lane & 15);
  const int koff = (lane < 16) ? 0 : 2;
  const float* ap = X + (size_t)row * ND + koff;
  #pragma unroll
  for (int s = 0; s < 32; ++s) areg[s] = *(const v2f*)(ap + s * 4);
}

// Cooperative copy of B-tile (rows j0..j0+15 of X) into padded LDS. 256 threads,
// 512 float4s -> 2 per thread. Caller handles barriers.
__device__ __forceinline__ void stage_btile(float* bs, const float* __restrict__ X,
                                            int j0, int tid) {
  #pragma unroll
  for (int r = 0; r < 2; ++r) {
    int idx = tid + r * 256;           // 0..511
    int row = idx >> 5;                // 32 float4 per row
    int c4  = (idx & 31) * 4;
    float4 v = *(const float4*)(X + (size_t)(j0 + row) * ND + c4);
    *(float4*)&bs[row * LDS_STRIDE + c4] = v;
  }
}

// 16x16 f32 dot tile: A from registers, B fragments from LDS (ds_load_b64).
__device__ __forceinline__ v8f tile_gemm_lds(const v2f* areg, const float* bs, int lane) {
  const int colr = lane & 15;
  const int koff = (lane < 16) ? 0 : 2;
  const float* bp = bs + colr * LDS_STRIDE + koff;
  v8f c = {};
  #pragma unroll
  for (int s = 0; s < 32; ++s) {
    v2f b = *(const v2f*)(bp + s * 4);
    c = __builtin_amdgcn_wmma_f32_16x16x4_f32(false, areg[s], false, b,
                                              (short)0, c, false, false);
  }
  return c;
}

// ---------------- Kernel 1: row squared norms + adaptive margins ----------------
__global__ void k_norms(const float* __restrict__ X, float* __restrict__ sq,
                        float* __restrict__ margin) {
  const int wave = (blockIdx.x * blockDim.x + threadIdx.x) >> 5;
  const int lane = threadIdx.x & 31;
  if (wave >= NB) return;
  const float4 v = *(const float4*)(X + (size_t)wave * ND + lane * 4);
  float s = v.x * v.x + v.y * v.y + v.z * v.z + v.w * v.w;
  s = wave_red_addf(s);
  if (lane == 0) {
    sq[wave] = s;
    float n = sqrtf(s);
    margin[wave] = fmaxf(1.0f * (1.0f + 2.0f * n), 0.1f);  // MARGIN=1, BF=2, THRESH=0.1
  }
}

// ---------------- Kernel 2: hardest positive, pos/neg counts, pos-dist sum ------
__global__ __launch_bounds__(256, 1)
void k_pos(const float* __restrict__ X, const int* __restrict__ labels,
           const float* __restrict__ sq,
           unsigned* __restrict__ hardest, unsigned* __restrict__ poscnt,
           unsigned* __restrict__ negcnt, float* __restrict__ scal) {
  __shared__ float bs[16 * LDS_STRIDE];
  const int tid  = threadIdx.x;
  const int lane = tid & 31;
  const int w    = tid >> 5;
  const int i0   = blockIdx.x * 128 + w * 16;   // this wave's M-tile
  const int hb   = (lane < 16) ? 0 : 8;         // C-row half offset

  v2f areg[32];
  load_atile(areg, X, i0, lane);

  float si[8]; int li[8];
  #pragma unroll
  for (int k = 0; k < 8; ++k) { int r = i0 + hb + k; si[k] = sq[r]; li[k] = labels[r]; }

  float pmax[8]; unsigned cnt[8];  // cnt: pos<<16 | neg (slice counts <= 512)
  float pdsum = 0.0f;
  #pragma unroll
  for (int k = 0; k < 8; ++k) { pmax[k] = 0.0f; cnt[k] = 0u; }

  const int tiles = (NB / 16) / gridDim.y;
  const int t0 = blockIdx.y * tiles;
  for (int t = 0; t < tiles; ++t) {
    const int j0 = (t0 + t) * 16;
    __syncthreads();                 // previous tile fully consumed
    stage_btile(bs, X, j0, tid);
    __syncthreads();                 // tile visible to all waves

    const int n  = j0 + (lane & 15);
    v8f c = tile_gemm_lds(areg, bs, lane);
    const float sj = sq[n];
    const int   lj = labels[n];
    #pragma unroll
    for (int k = 0; k < 8; ++k) {
      const int row = i0 + hb + k;
      float d = pdist(c[k], si[k], sj);
      if (li[k] == lj) {
        if (n != row) { cnt[k] += 0x10000u; pdsum += d; pmax[k] = fmaxf(pmax[k], d); }
      } else {
        cnt[k] += 1u;
      }
    }
  }

  // per-row reduce across the 16-lane half
  #pragma unroll
  for (int k = 0; k < 8; ++k) {
    #pragma unroll
    for (int m = 1; m < 16; m <<= 1) {
      pmax[k] = fmaxf(pmax[k], __shfl_xor(pmax[k], m, 32));
      cnt[k] += __shfl_xor(cnt[k], m, 32);
    }
  }
  pdsum = wave_red_addf(pdsum);
  if ((lane & 15) == 0) {
    #pragma unroll
    for (int k = 0; k < 8; ++k) {
      int row = i0 + hb + k;
      atomicMax(&hardest[row], __float_as_uint(pmax[k]));  // d >= 0: uint order == float order
      atomicAdd(&poscnt[row], cnt[k] >> 16);
      atomicAdd(&negcnt[row], cnt[k] & 0xFFFFu);
    }
  }
  if (lane == 0) atomicAdd(&scal[2], pdsum);
}

// ---------------- Kernel 3: hinge loss sum + active count ----------------------
__global__ __launch_bounds__(256, 1)
void k_loss(const float* __restrict__ X, const int* __restrict__ labels,
            const float* __restrict__ sq, const float* __restrict__ margin,
            const unsigned* __restrict__ hardest,
            const unsigned* __restrict__ poscnt,
            const unsigned* __restrict__ negcnt, float* __restrict__ scal) {
  __shared__ float bs[16 * LDS_STRIDE];
  const int tid  = threadIdx.x;
  const int lane = tid & 31;
  const int w    = tid >> 5;
  const int i0   = blockIdx.x * 128 + w * 16;
  const int hb   = (lane < 16) ? 0 : 8;

  v2f areg[32];
  load_atile(areg, X, i0, lane);

  float si[8], hpm[8];  // hpm = hardest_pos + margin (precombined)
  int li[8]; bool valid[8];
  #pragma unroll
  for (int k = 0; k < 8; ++k) {
    int r = i0 + hb + k;
    si[k] = sq[r]; li[k] = labels[r];
    hpm[k] = __uint_as_float(hardest[r]) + margin[r];
    valid[k] = (poscnt[r] > 0u) && (negcnt[r] > 0u);
  }

  float lsum = 0.0f; unsigned nact = 0u;
  const int tiles = (NB / 16) / gridDim.y;
  const int t0 = blockIdx.y * tiles;
  for (int t = 0; t < tiles; ++t) {
    const int j0 = (t0 + t) * 16;
    __syncthreads();
    stage_btile(bs, X, j0, tid);
    __syncthreads();

    const int n  = j0 + (lane & 15);
    v8f c = tile_gemm_lds(areg, bs, lane);
    const float sj = sq[n];
    const int   lj = labels[n];
    #pragma unroll
    for (int k = 0; k < 8; ++k) {
      if (valid[k] && (li[k] != lj)) {     // diagonal has equal labels -> excluded
        float d   = pdist(c[k], si[k], sj);
        float raw = hpm[k] - d;
        if (raw > 0.0f) { lsum += raw; nact++; }
      }
    }
  }
  lsum = wave_red_addf(lsum);
  nact = wave_red_addu(nact);
  if (lane == 0) {
    atomicAdd(&scal[0], lsum);
    atomicAdd((unsigned*)&scal[1], nact);
  }
}

// ---------------- Kernel 4: finalize the 5 outputs -----------------------------
__global__ void k_final(const unsigned* __restrict__ poscnt,
                        const unsigned* __restrict__ negcnt,
                        const float* __restrict__ scal, float* __restrict__ out) {
  __shared__ unsigned s_tt[256], s_pc[256];
  const int tid = threadIdx.x;
  unsigned tt = 0, pc = 0;
  for (int r = tid; r < NB; r += 256) {
    unsigned p = poscnt[r], nn = negcnt[r];
    pc += p;
    if (p > 0u && nn > 0u) tt += nn;   // total_triplets = sum over valid anchors of #neg
  }
  s_tt[tid] = tt; s_pc[tid] = pc;
  __syncthreads();
  for (int s = 128; s > 0; s >>= 1) {
    if (tid < s) { s_tt[tid] += s_tt[tid + s]; s_pc[tid] += s_pc[tid + s]; }
    __syncthreads();
  }
  if (tid == 0) {
    unsigned total = s_tt[0], posct = s_pc[0];
    float losssum  = scal[0];
    unsigned numact = ((const unsigned*)scal)[1];
    float pdsum    = scal[2];
    float denom    = (float)(total > 0u ? total : 1u);
    out[0] = losssum / denom;                               // loss
    out[1] = (float)numact;                                 // num_active
    out[2] = (float)total;                                  // total_triplets
    out[3] = (float)numact / fmaxf((float)total, 1.0f);     // active_ratio
    out[4] = (posct > 0u) ? pdsum / (float)posct : 0.0f;    // mean_distance
  }
}

extern "C" void kernel_launch(void* const* d_in, const int* in_sizes, int n_in,
                              void* d_out, int out_size, void* d_ws, size_t ws_size,
                              hipStream_t stream) {
  const float* X      = (const float*)d_in[0];
  const int*   labels = (const int*)d_in[1];

  float*    ws      = (float*)d_ws;
  float*    sq      = ws;                       // [NB]
  float*    margin  = ws + NB;                  // [NB]
  unsigned* hardest = (unsigned*)(ws + 2 * NB); // [NB] f32 bits, init 0
  unsigned* poscnt  = (unsigned*)(ws + 3 * NB); // [NB]
  unsigned* negcnt  = (unsigned*)(ws + 4 * NB); // [NB]
  float*    scal    = ws + 5 * NB;              // [0]=losssum [1]=numactive [2]=posdistsum

  // zero hardest/poscnt/negcnt/scalars (deterministic re-init every call)
  hipMemsetAsync(hardest, 0, (size_t)(3 * NB + 8) * sizeof(float), stream);

  k_norms<<<NB / 8, 256, 0, stream>>>(X, sq, margin);

  dim3 grid(NB / 128, 8);   // 32 M-supertiles (8 waves x 16 rows) x 8 column slices
  k_pos <<<grid, 256, 0, stream>>>(X, labels, sq, hardest, poscnt, negcnt, scal);
  k_loss<<<grid, 256, 0, stream>>>(X, labels, sq, margin, hardest, poscnt, negcnt, scal);
  k_final<<<1, 256, 0, stream>>>(poscnt, negcnt, scal, (float*)d_out);
}